// RNNLM_49125835931967
// MI455X (gfx1250) — compile-verified
//
#include <hip/hip_runtime.h>
#include <hip/hip_bf16.h>
#include <math.h>

// Problem constants (reference: S=128, B=64, V=32000, E=32, H=16)
#define SEQ   128
#define BAT   64
#define VOC   32000
#define EMB   32
#define HID   16
#define ROWS  (SEQ * BAT)       // 8192 flattened (s,b) rows
#define NCHUNK 256              // columns of Wo staged in LDS per step
#define NSPLIT 5                // vocab split for parallelism: 5 * 25 * 256 = 32000
#define CHUNKS_PER_SPLIT 25

typedef __attribute__((ext_vector_type(2))) float v2f;
typedef __attribute__((ext_vector_type(8))) float v8f;
typedef __attribute__((ext_vector_type(4))) unsigned int v4u;
typedef __attribute__((ext_vector_type(8))) int v8i32;
typedef __attribute__((ext_vector_type(4))) int v4i32;

// ---------------------------------------------------------------------------
// K1: embedding gather + X @ Wx  ->  XW [ROWS, HID]
// ---------------------------------------------------------------------------
__global__ __launch_bounds__(256)
void embed_xw_kernel(const int* __restrict__ ids,
                     const float* __restrict__ lookup,
                     const float* __restrict__ Wx,
                     float* __restrict__ XW) {
  int gid = blockIdx.x * blockDim.x + threadIdx.x;   // ROWS*HID threads
  int row = gid >> 4;
  int j   = gid & 15;
  int id  = ids[row];
  const float* xrow = lookup + (long)id * EMB;
  float acc = 0.f;
#pragma unroll
  for (int e = 0; e < EMB; ++e)
    acc = fmaf(xrow[e], Wx[e * HID + j], acc);
  XW[row * HID + j] = acc;
}

// ---------------------------------------------------------------------------
// K2: serial Elman recurrence  h = tanh(XW_t + h @ Wh), writes Hall [ROWS,HID]
// ---------------------------------------------------------------------------
__global__ __launch_bounds__(1024)
void recurrence_kernel(const float* __restrict__ XW,
                       const float* __restrict__ Wh,
                       const float* __restrict__ h0,
                       float* __restrict__ Hall) {
  __shared__ float hs[BAT * HID];    // 1024 floats
  __shared__ float whs[HID * HID];   // 256 floats
  int t = threadIdx.x;               // 0..1023
  if (t < HID * HID) whs[t] = Wh[t];
  hs[t] = h0[t];
  __syncthreads();
  int b = t >> 4, j = t & 15;
  for (int s = 0; s < SEQ; ++s) {
    float acc = XW[s * (BAT * HID) + t];
#pragma unroll
    for (int k = 0; k < HID; ++k)
      acc = fmaf(hs[b * HID + k], whs[k * HID + j], acc);
    float hn = tanhf(acc);
    __syncthreads();
    hs[t] = hn;
    Hall[s * (BAT * HID) + t] = hn;
    __syncthreads();
  }
}

// ---------------------------------------------------------------------------
// TDM: async DMA of one 16 x NCHUNK fp32 tile of Wo (row stride VOC) into LDS.
// D# built per CDNA5 ISA ch.8: group0 = {count, lds_addr, global_addr, type=2},
// group1 = {data_size=4B, tensor_dim0=VOC, tensor_dim1=16, tile_dim0=NCHUNK,
//           tile_dim1=16, tensor_dim0_stride=VOC}. Groups 2/3 zero (2D tile).
// This toolchain exposes the 6-arg builtin:
//   (uint32x4 g0, int32x8 g1, int32x4, int32x4, int32x8, i32 cpol)
// ---------------------------------------------------------------------------
__device__ __forceinline__ void tdm_load_chunk(const float* gsrc, void* lds_dst) {
  unsigned long long ga = (unsigned long long)(uintptr_t)gsrc;
  unsigned int la = (unsigned int)(uintptr_t)lds_dst;
  v4u g0;
  g0[0] = 1u;                                       // count=1 (valid), user mode
  g0[1] = la;                                       // lds_addr (bytes)
  g0[2] = (unsigned int)ga;                         // global_addr[31:0]
  g0[3] = (unsigned int)(ga >> 32) | (2u << 30);    // global_addr[56:32] | type=2
  v8i32 g1;
  g1[0] = 0x00020000;                               // workgroup_mask=0, data_size=2(4B)
  g1[1] = (int)((VOC & 0xffff) << 16);              // tensor_dim0[15:0] (VOC<64K)
  g1[2] = (int)((VOC >> 16) | (HID << 16));         // tensor_dim0 hi | tensor_dim1 lo
  g1[3] = (int)(NCHUNK << 16);                      // tensor_dim1 hi | tile_dim0
  g1[4] = HID;                                      // tile_dim1=16, tile_dim2=0
  g1[5] = VOC;                                      // tensor_dim0_stride[31:0]
  g1[6] = 0;                                        // stride hi | dim1_stride lo
  g1[7] = 0;
  v4i32 z4 = {0, 0, 0, 0};
  v8i32 z8 = {0, 0, 0, 0, 0, 0, 0, 0};
  __builtin_amdgcn_tensor_load_to_lds(g0, g1, z4, z4, z8, 0);
}

// ---------------------------------------------------------------------------
// Output GEMM machinery (Hall[ROWS,16] x Wo[16,VOC]) via V_WMMA_F32_16X16X4:
// K=16 -> 4 chained k-steps. Block = 128 threads = 4 waves; wave w owns rows
// [mbase, mbase+16). B tiles staged in double-buffered LDS by the TDM.
// ---------------------------------------------------------------------------
__device__ __forceinline__ void load_a_frags(const float* __restrict__ Hall,
                                             int mbase, int l16, int khalf,
                                             v2f afrag[4]) {
  const float* arow = Hall + (long)(mbase + l16) * HID;
#pragma unroll
  for (int kk = 0; kk < 4; ++kk) {
    afrag[kk].x = arow[kk * 4 + khalf];
    afrag[kk].y = arow[kk * 4 + khalf + 1];
  }
}

__device__ __forceinline__ v8f tile_wmma(const float* __restrict__ Bs,
                                         const v2f afrag[4],
                                         int khalf, int ncol) {
  v8f acc = {};
#pragma unroll
  for (int kk = 0; kk < 4; ++kk) {
    v2f bf;
    bf.x = Bs[(kk * 4 + khalf) * NCHUNK + ncol];
    bf.y = Bs[(kk * 4 + khalf + 1) * NCHUNK + ncol];
    acc = __builtin_amdgcn_wmma_f32_16x16x4_f32(false, afrag[kk],
                                                false, bf,
                                                (short)0, acc,
                                                false, false);
  }
  return acc;
}

// ---------------------------------------------------------------------------
// K3: partial log-softmax stats per (64-row block, vocab split).
// ---------------------------------------------------------------------------
__global__ __launch_bounds__(128)
void logits_stats_kernel(const float* __restrict__ Hall,
                         const float* __restrict__ Wo,
                         float* __restrict__ stats) {
  __shared__ __align__(16) float Bs[2][HID * NCHUNK];  // 2 x 16 KB
  const int tid  = threadIdx.x;
  const int lane = tid & 31;
  const int wave = tid >> 5;
  const int hf   = lane >> 4;       // half-wave id
  const int l16  = lane & 15;
  const int khalf = hf * 2;
  const int mbase = blockIdx.x * 64 + wave * 16;
  const int ns    = blockIdx.y;

  v2f afrag[4];
  load_a_frags(Hall, mbase, l16, khalf, afrag);

  float m[8], ssum[8];
#pragma unroll
  for (int r = 0; r < 8; ++r) { m[r] = -3.402823466e38f; ssum[r] = 0.f; }

  const int c0 = ns * CHUNKS_PER_SPLIT;
  const int cend = c0 + CHUNKS_PER_SPLIT;
  if (wave == 0) {
    tdm_load_chunk(Wo + (long)c0 * NCHUNK, &Bs[0][0]);
    __builtin_amdgcn_s_wait_tensorcnt(0);
  }
  __syncthreads();

  for (int c = c0; c < cend; ++c) {
    const int p = (c - c0) & 1;
    if (wave == 0 && c + 1 < cend)              // prefetch next chunk via TDM
      tdm_load_chunk(Wo + (long)(c + 1) * NCHUNK, &Bs[p ^ 1][0]);
    if (c + 2 < cend)                           // warm L2 two chunks ahead
      __builtin_prefetch(Wo + (long)(tid >> 3) * VOC + (long)(c + 2) * NCHUNK +
                         ((tid & 7) << 5), 0, 0);
    const float* bs = &Bs[p][0];
#pragma unroll 1
    for (int nt = 0; nt < NCHUNK / 16; ++nt) {
      v8f acc = tile_wmma(bs, afrag, khalf, nt * 16 + l16);
#pragma unroll
      for (int r = 0; r < 8; ++r) {
        float v = acc[r];
        if (v > m[r]) { ssum[r] *= __expf(m[r] - v); m[r] = v; }
        ssum[r] += __expf(v - m[r]);
      }
    }
    __syncthreads();                            // all waves done with Bs[p]
    if (wave == 0) __builtin_amdgcn_s_wait_tensorcnt(0);
    __syncthreads();                            // next chunk visible to all
  }

  // reduce (m, s) across the 16 lanes of each half-wave
#pragma unroll
  for (int r = 0; r < 8; ++r) {
    float mr = m[r], sr = ssum[r];
#pragma unroll
    for (int off = 1; off < 16; off <<= 1) {
      float mo = __shfl_xor(mr, off, 32);
      float so = __shfl_xor(sr, off, 32);
      float mn = fmaxf(mr, mo);
      sr = sr * __expf(mr - mn) + so * __expf(mo - mn);
      mr = mn;
    }
    m[r] = mr; ssum[r] = sr;
  }
  if (l16 == 0) {
#pragma unroll
    for (int r = 0; r < 8; ++r) {
      int rowg = mbase + r + hf * 8;
      stats[((long)rowg * NSPLIT + ns) * 2 + 0] = m[r];
      stats[((long)rowg * NSPLIT + ns) * 2 + 1] = ssum[r];
    }
  }
}

// ---------------------------------------------------------------------------
// K4: merge NSPLIT partial stats -> logZ[row]
// ---------------------------------------------------------------------------
__global__ __launch_bounds__(256)
void merge_stats_kernel(const float* __restrict__ stats,
                        float* __restrict__ logZ) {
  int row = blockIdx.x * blockDim.x + threadIdx.x;   // ROWS threads
  float M = -3.402823466e38f;
#pragma unroll
  for (int ns = 0; ns < NSPLIT; ++ns)
    M = fmaxf(M, stats[((long)row * NSPLIT + ns) * 2 + 0]);
  float S = 0.f;
#pragma unroll
  for (int ns = 0; ns < NSPLIT; ++ns) {
    float mi = stats[((long)row * NSPLIT + ns) * 2 + 0];
    float si = stats[((long)row * NSPLIT + ns) * 2 + 1];
    S += si * __expf(mi - M);
  }
  logZ[row] = M + __logf(S);
}

// ---------------------------------------------------------------------------
// K5: recompute logits, store logit - logZ  (single 1.05 GB store pass)
// ---------------------------------------------------------------------------
__global__ __launch_bounds__(128)
void logits_store_kernel(const float* __restrict__ Hall,
                         const float* __restrict__ Wo,
                         const float* __restrict__ logZ,
                         float* __restrict__ out) {
  __shared__ __align__(16) float Bs[2][HID * NCHUNK];
  const int tid  = threadIdx.x;
  const int lane = tid & 31;
  const int wave = tid >> 5;
  const int hf   = lane >> 4;
  const int l16  = lane & 15;
  const int khalf = hf * 2;
  const int mbase = blockIdx.x * 64 + wave * 16;
  const int ns    = blockIdx.y;

  v2f afrag[4];
  load_a_frags(Hall, mbase, l16, khalf, afrag);

  float lz[8];
#pragma unroll
  for (int r = 0; r < 8; ++r)
    lz[r] = logZ[mbase + r + hf * 8];

  const int c0 = ns * CHUNKS_PER_SPLIT;
  const int cend = c0 + CHUNKS_PER_SPLIT;
  if (wave == 0) {
    tdm_load_chunk(Wo + (long)c0 * NCHUNK, &Bs[0][0]);
    __builtin_amdgcn_s_wait_tensorcnt(0);
  }
  __syncthreads();

  for (int c = c0; c < cend; ++c) {
    const int p = (c - c0) & 1;
    if (wave == 0 && c + 1 < cend)
      tdm_load_chunk(Wo + (long)(c + 1) * NCHUNK, &Bs[p ^ 1][0]);
    if (c + 2 < cend)
      __builtin_prefetch(Wo + (long)(tid >> 3) * VOC + (long)(c + 2) * NCHUNK +
                         ((tid & 7) << 5), 0, 0);
    const float* bs = &Bs[p][0];
    const int ncb = c * NCHUNK;
#pragma unroll 1
    for (int nt = 0; nt < NCHUNK / 16; ++nt) {
      v8f acc = tile_wmma(bs, afrag, khalf, nt * 16 + l16);
      const int col = ncb + nt * 16 + l16;
#pragma unroll
      for (int r = 0; r < 8; ++r) {
        int rowg = mbase + r + hf * 8;
        out[(long)rowg * VOC + col] = acc[r] - lz[r];
      }
    }
    __syncthreads();
    if (wave == 0) __builtin_amdgcn_s_wait_tensorcnt(0);
    __syncthreads();
  }
}

// ---------------------------------------------------------------------------
// Host entry
// ---------------------------------------------------------------------------
extern "C" void kernel_launch(void* const* d_in, const int* in_sizes, int n_in,
                              void* d_out, int out_size, void* d_ws, size_t ws_size,
                              hipStream_t stream) {
  const int*   ids    = (const int*)  d_in[0];   // [S,B]
  const float* lookup = (const float*)d_in[1];   // [V,E]
  const float* Wx     = (const float*)d_in[2];   // [E,H]
  const float* Wh     = (const float*)d_in[3];   // [H,H]
  const float* Wo     = (const float*)d_in[4];   // [H,V]
  const float* h0     = (const float*)d_in[5];   // [B,H]
  float* out = (float*)d_out;                    // [S,B,V]

  float* ws    = (float*)d_ws;
  float* XW    = ws;                             // ROWS*HID      = 131072
  float* Hall  = XW    + ROWS * HID;             // ROWS*HID      = 131072
  float* stats = Hall  + ROWS * HID;             // ROWS*NSPLIT*2 =  81920
  float* logZ  = stats + ROWS * NSPLIT * 2;      // ROWS          =   8192

  embed_xw_kernel<<<(ROWS * HID) / 256, 256, 0, stream>>>(ids, lookup, Wx, XW);
  recurrence_kernel<<<1, BAT * HID, 0, stream>>>(XW, Wh, h0, Hall);

  dim3 ggrid(ROWS / 64, NSPLIT);
  logits_stats_kernel<<<ggrid, 128, 0, stream>>>(Hall, Wo, stats);
  merge_stats_kernel<<<ROWS / 256, 256, 0, stream>>>(stats, logZ);
  logits_store_kernel<<<ggrid, 128, 0, stream>>>(Hall, Wo, logZ, out);
}